// FullRankLinearKernel_86766929314329
// MI455X (gfx1250) — compile-verified
//
#include <hip/hip_runtime.h>
#include <hip/hip_bf16.h>

// Problem constants from the reference
#define B_  8
#define CI_ 32
#define CO_ 64
#define F_  4
#define H_  64
#define W_  64
#define HW_ (H_ * W_)

typedef __attribute__((ext_vector_type(2))) float v2f;
typedef __attribute__((ext_vector_type(8))) float v8f;

// ---------------------------------------------------------------------------
// Kernel A: trapezoid-weighted reductions over the (h,w) plane.
//   iv[b,i]      = sum_{h,w} wgt * v[b,i,h,w]
//   T[b,i*2+d]   = sum_{h,w} wgt * vc[h,w,d] * v[b,i,h,w]
// One block per (b,i) plane (256 blocks), 256 threads. B128 loads (4 pixels /
// thread / iter), wave32 shuffle reduction. Reads v exactly once (4 MiB).
// ---------------------------------------------------------------------------
__global__ void frl_reduce_kernel(const float* __restrict__ v,
                                  const float* __restrict__ vc,
                                  float* __restrict__ iv,
                                  float* __restrict__ T) {
    const int plane = blockIdx.x;          // b*CI + i
    const int t = threadIdx.x;
    const float* vp = v + (size_t)plane * HW_;

    const float dx = vc[2] - vc[0];        // v_coords[0,1,0] - v_coords[0,0,0]
    const float scale = dx * dx;

    float a0 = 0.f, a1 = 0.f, a2 = 0.f;
    // 4 consecutive pixels per thread -> global_load_b128; a quad never
    // crosses a row (W=64), so only elements 0 and 3 can touch w-edges.
    for (int p = t * 4; p < HW_; p += 1024) {
        const int h = p >> 6, w = p & 63;
        const float ch  = (h == 0 || h == H_ - 1) ? 0.5f : 1.0f;
        const float cw0 = (w == 0) ? 0.5f : 1.0f;
        const float cw3 = (w + 3 == W_ - 1) ? 0.5f : 1.0f;
        const float wgt0 = ch * cw0 * scale;
        const float wgt1 = ch * scale;          // interior elements 1,2
        const float wgt3 = ch * cw3 * scale;

        const float4 vv = *(const float4*)(vp + p);
        const float4 c0 = *(const float4*)(vc + 2 * p);      // gx0 gy0 gx1 gy1
        const float4 c1 = *(const float4*)(vc + 2 * p + 4);  // gx2 gy2 gx3 gy3

        const float w0 = wgt0 * vv.x;
        const float w1 = wgt1 * vv.y;
        const float w2 = wgt1 * vv.z;
        const float w3 = wgt3 * vv.w;
        a0 += w0 + w1 + w2 + w3;
        a1 += w0 * c0.x + w1 * c0.z + w2 * c1.x + w3 * c1.z;
        a2 += w0 * c0.y + w1 * c0.w + w2 * c1.y + w3 * c1.w;
    }

    // wave32 shuffle reduction, then combine the 8 wave partials.
    #pragma unroll
    for (int m = 16; m >= 1; m >>= 1) {
        a0 += __shfl_xor(a0, m, 32);
        a1 += __shfl_xor(a1, m, 32);
        a2 += __shfl_xor(a2, m, 32);
    }
    __shared__ float r0[8], r1[8], r2[8];
    const int wave = t >> 5, lane = t & 31;
    if (lane == 0) { r0[wave] = a0; r1[wave] = a1; r2[wave] = a2; }
    __syncthreads();
    if (t == 0) {
        float s0 = 0.f, s1 = 0.f, s2 = 0.f;
        #pragma unroll
        for (int k = 0; k < 8; ++k) { s0 += r0[k]; s1 += r1[k]; s2 += r2[k]; }
        iv[plane]        = s0;
        T[2 * plane]     = s1;      // d = 0 (x)
        T[2 * plane + 1] = s2;      // d = 1 (y)
    }
}

// ---------------------------------------------------------------------------
// Kernel B: the GEMM core on the CDNA5 matrix pipe (fp32 WMMA 16x16x4).
//   WxS[o,i,d] = sum_f Wx[f,o,i,d];  WyS likewise  (staged in LDS)
//   Mx[b,o] = sum_i iv[b,i]  * WxS[o,i,0]          (K=32)
//   My[b,o] = sum_i iv[b,i]  * WxS[o,i,1]          (K=32)
//   S [b,o] = sum_k T[b,k]   * WyS_k[o,k], k=i*2+d (K=64)
// One workgroup, 4 waves; wave w owns CO columns [16w, 16w+16).
// A operands (iv, T) are staged into LDS zero-padded to 16 rows so the inner
// WMMA loops are pure ds_load_b64 + v_wmma with no EXEC predication.
// WMMA layout (CDNA5 ISA): A 16x4 f32 -> 2 VGPRs, lanes 0-15 K={0,1},
// lanes 16-31 K={2,3}; B 4x16 column-striped (lane = N mod 16, half selects
// K pair); C/D 8 VGPRs, lane = N, vgpr r = row M=r (+8 upper half).
// ---------------------------------------------------------------------------
__global__ void frl_gemm_kernel(const float* __restrict__ Wx,
                                const float* __restrict__ Wy,
                                const float* __restrict__ iv,
                                const float* __restrict__ T,
                                float* __restrict__ Mx,
                                float* __restrict__ My,
                                float* __restrict__ S) {
    __shared__ float wxs[2 * CO_ * CI_];   // [d][o][i]          16 KB
    __shared__ float wys[CO_ * 2 * CI_];   // [o][i*2+d]         16 KB
    __shared__ float ivs[16 * CI_];        // [row16][i]          2 KB (rows 8..15 = 0)
    __shared__ float Ts [16 * 2 * CI_];    // [row16][i*2+d]      4 KB (rows 8..15 = 0)
    const int t = threadIdx.x;             // 0..127

    // Fold the f-sum while staging the weights into LDS.
    for (int idx = t; idx < 2 * CO_ * CI_; idx += 128) {
        const int d = idx >> 11, rem = idx & 2047;
        const int o = rem >> 5, i = rem & 31;
        float s = 0.f;
        #pragma unroll
        for (int f = 0; f < F_; ++f)
            s += Wx[((f * CO_ + o) * CI_ + i) * 2 + d];
        wxs[idx] = s;                      // idx == d*2048 + o*32 + i
    }
    for (int idx = t; idx < CO_ * 2 * CI_; idx += 128) {
        const int o = idx >> 6, k = idx & 63;
        const int i = k >> 1, d = k & 1;
        float s = 0.f;
        #pragma unroll
        for (int f = 0; f < F_; ++f)
            s += Wy[((f * CO_ + o) * CI_ + i) * 2 + d];
        wys[idx] = s;                      // idx == o*64 + k
    }
    // Zero-padded A operands (b rows 8..15 are padding).
    for (int idx = t; idx < 16 * CI_; idx += 128) {
        const int r = idx >> 5;
        ivs[idx] = (r < B_) ? iv[idx] : 0.f;       // iv is [8][32] flat
    }
    for (int idx = t; idx < 16 * 2 * CI_; idx += 128) {
        const int r = idx >> 6;
        Ts[idx] = (r < B_) ? T[idx] : 0.f;         // T is [8][64] flat
    }
    __syncthreads();

    const int wave = t >> 5;               // CO tile id (wave-uniform)
    const int lane = t & 31;
    const int row  = lane & 15;            // M index within lane half (= b)
    const int kk   = (lane >> 4) << 1;     // 0 for lanes 0-15, 2 for 16-31
    const int ocol = wave * 16 + row;      // B-operand column (N = lane&15)

    v8f accx = {}, accy = {}, accs = {};

    // Mx: D = iv[16x32] * WxS_d0[32x16tile]
    #pragma unroll
    for (int step = 0; step < 8; ++step) {
        const int k0 = step * 4 + kk;      // k0 even -> 8B-aligned ds_load_b64
        const v2f a = *(const v2f*)(ivs + row * CI_ + k0);
        const v2f b = *(const v2f*)(wxs + 0 * 2048 + ocol * CI_ + k0);
        accx = __builtin_amdgcn_wmma_f32_16x16x4_f32(
            false, a, false, b, (short)0, accx, false, false);
    }
    // My: same with d = 1
    #pragma unroll
    for (int step = 0; step < 8; ++step) {
        const int k0 = step * 4 + kk;
        const v2f a = *(const v2f*)(ivs + row * CI_ + k0);
        const v2f b = *(const v2f*)(wxs + 1 * 2048 + ocol * CI_ + k0);
        accy = __builtin_amdgcn_wmma_f32_16x16x4_f32(
            false, a, false, b, (short)0, accy, false, false);
    }
    // S: D = T[16x64] * WyS[64x16tile], k = i*2+d interleaved
    #pragma unroll
    for (int step = 0; step < 16; ++step) {
        const int k0 = step * 4 + kk;
        const v2f a = *(const v2f*)(Ts + row * 64 + k0);
        const v2f b = *(const v2f*)(wys + ocol * 64 + k0);
        accs = __builtin_amdgcn_wmma_f32_16x16x4_f32(
            false, a, false, b, (short)0, accs, false, false);
    }

    // D layout: lane (=N) in 0..15 holds rows M=0..7 in vgprs 0..7 -> b=0..7.
    if (lane < 16) {
        const int o = wave * 16 + lane;
        #pragma unroll
        for (int r = 0; r < B_; ++r) {
            Mx[r * CO_ + o] = accx[r];
            My[r * CO_ + o] = accy[r];
            S [r * CO_ + o] = accs[r];
        }
    }
}

// ---------------------------------------------------------------------------
// Kernel C: streaming output (8 MiB of writes — the bandwidth bottleneck).
//   out[b,o,h,w] = Mx[b,o]*gx + My[b,o]*gy + S[b,o]
// One block per (b,o) pair; coefficients are block-uniform. B128 I/O.
// ---------------------------------------------------------------------------
__global__ void frl_out_kernel(const float* __restrict__ tc,
                               const float* __restrict__ Mx,
                               const float* __restrict__ My,
                               const float* __restrict__ S,
                               float* __restrict__ out) {
    const int bo = blockIdx.x;             // b*CO + o
    const int t = threadIdx.x;
    const float mx = Mx[bo], my = My[bo], s = S[bo];
    float* op = out + (size_t)bo * HW_;
    for (int p = t * 4; p < HW_; p += 1024) {
        const float4 c0 = *(const float4*)(tc + 2 * p);      // gx0 gy0 gx1 gy1
        const float4 c1 = *(const float4*)(tc + 2 * p + 4);  // gx2 gy2 gx3 gy3
        float4 o;
        o.x = fmaf(mx, c0.x, fmaf(my, c0.y, s));
        o.y = fmaf(mx, c0.z, fmaf(my, c0.w, s));
        o.z = fmaf(mx, c1.x, fmaf(my, c1.y, s));
        o.w = fmaf(mx, c1.z, fmaf(my, c1.w, s));
        *(float4*)(op + p) = o;                              // global_store_b128
    }
}

extern "C" void kernel_launch(void* const* d_in, const int* in_sizes, int n_in,
                              void* d_out, int out_size, void* d_ws, size_t ws_size,
                              hipStream_t stream) {
    const float* v  = (const float*)d_in[0];  // [B,CI,H,W]
    const float* vc = (const float*)d_in[1];  // [H,W,2]
    const float* tc = (const float*)d_in[2];  // [H,W,2]
    const float* Wx = (const float*)d_in[3];  // [F,CO,CI,2]
    const float* Wy = (const float*)d_in[4];  // [F,CO,CI,2]
    float* out = (float*)d_out;               // [B,CO,H,W]

    float* ws = (float*)d_ws;
    float* iv = ws;            // 256
    float* T  = ws + 256;      // 512  (b, i, d interleaved)
    float* Mx = ws + 768;      // 512
    float* My = ws + 1280;     // 512
    float* Sv = ws + 1792;     // 512

    frl_reduce_kernel<<<B_ * CI_, 256, 0, stream>>>(v, vc, iv, T);
    frl_gemm_kernel<<<1, 128, 0, stream>>>(Wx, Wy, iv, T, Mx, My, Sv);
    frl_out_kernel<<<B_ * CO_, 256, 0, stream>>>(tc, Mx, My, Sv, out);
}